// GraphAttentionLayer_33028298506675
// MI455X (gfx1250) — compile-verified
//
#include <hip/hip_runtime.h>
#include <hip/hip_bf16.h>

typedef __attribute__((ext_vector_type(2))) float v2f;
typedef __attribute__((ext_vector_type(8))) float v8f;

#define B_DIM  16
#define N_DIM  2048
#define F_INN  128
#define F_OUTT 64

// ---------------------------------------------------------------------------
// Kernel 1: h = X @ W   (B*N x 128) @ (128 x 64), fp32 WMMA 16x16x4
// block = 128 threads (4 waves); block owns 16 rows, wave w owns cols 16w..16w+15
// ---------------------------------------------------------------------------
__global__ __launch_bounds__(128) void gat_h_kernel(const float* __restrict__ X,
                                                    const float* __restrict__ W,
                                                    float* __restrict__ h) {
    const int tile = blockIdx.x;            // row tile over B*N (16 rows each)
    const int wave = threadIdx.x >> 5;      // 0..3  -> col tile
    const int lane = threadIdx.x & 31;
    const int m    = lane & 15;             // M row within tile
    const int kh   = (lane >> 4) * 2;       // K half-select: 0 or 2
    const int col  = wave * 16 + m;

    const float* __restrict__ xrow = X + (size_t)(tile * 16 + m) * F_INN;

    v8f acc = {};
    #pragma unroll 8
    for (int k = 0; k < F_INN; k += 4) {
        v2f a, b;
        // A 16x4 f32 layout: lanes 0-15 K={k,k+1}, lanes 16-31 K={k+2,k+3}
        a.x = xrow[k + kh];
        a.y = xrow[k + kh + 1];
        // B 4x16 f32 layout (row striped across lanes, K-halves across lane halves)
        b.x = W[(size_t)(k + kh)     * F_OUTT + col];
        b.y = W[(size_t)(k + kh + 1) * F_OUTT + col];
        acc = __builtin_amdgcn_wmma_f32_16x16x4_f32(false, a, false, b,
                                                    (short)0, acc, false, false);
    }

    // C/D 16x16 f32 layout: VGPR r -> M = r + 8*(lane>>4), N = lane&15
    float* hp = h + (size_t)(tile * 16) * F_OUTT + col;
    #pragma unroll
    for (int r = 0; r < 8; ++r) {
        const int rr = r + 8 * (lane >> 4);
        hp[(size_t)rr * F_OUTT] = acc[r];
    }
}

// ---------------------------------------------------------------------------
// Kernel 2: s1[i] = h[i,:] . a[:64],  s2[i] = h[i,:] . a[64:]
// ---------------------------------------------------------------------------
__global__ __launch_bounds__(256) void gat_scores_kernel(const float* __restrict__ h,
                                                         const float* __restrict__ a,
                                                         float* __restrict__ s1,
                                                         float* __restrict__ s2) {
    const int i = blockIdx.x * blockDim.x + threadIdx.x;
    if (i >= B_DIM * N_DIM) return;
    const float* hr = h + (size_t)i * F_OUTT;
    float acc1 = 0.f, acc2 = 0.f;
    #pragma unroll 8
    for (int f = 0; f < F_OUTT; ++f) {
        acc1 += hr[f] * a[f];
        acc2 += hr[f] * a[F_OUTT + f];
    }
    s1[i] = acc1;
    s2[i] = acc2;
}

// ---------------------------------------------------------------------------
// Kernel 3: fused masked-softmax attention + h' = att @ h + ELU
// grid = (N/16, B), block = 128 (4 waves). Block owns 16 rows x 64 out cols.
// adj (the 256 MB bandwidth driver) is streamed once with NT temporal hints
// (protects L2 residency of the 8 MB h matrix) and prefetched one chunk ahead
// (global_prefetch_b8). h chunk staged in LDS; unnormalized exp-weights
// accumulated via fp32 WMMA; normalized at the end. relu() bounds scores so
// softmax without max-subtraction is exact.
// ---------------------------------------------------------------------------
#define JB 128          // j-chunk width
#define ATT_LD 132      // padded LDS row stride (bank-conflict-free A fetch)

__global__ __launch_bounds__(128) void gat_attn_kernel(const float* __restrict__ h,
                                                       const float* __restrict__ s1,
                                                       const float* __restrict__ s2,
                                                       const int* __restrict__ adj,
                                                       float* __restrict__ out) {
    __shared__ float att_s[16 * ATT_LD];       // p = exp(relu(s1+s2)) or 0
    __shared__ float h_s[JB * F_OUTT];         // h chunk (j rows x 64 cols)
    __shared__ float rowsum_s[16];

    const int b    = blockIdx.y;
    const int tile = blockIdx.x;
    const int t    = threadIdx.x;
    const int wave = t >> 5;
    const int lane = t & 31;
    const int m    = lane & 15;
    const int kh   = (lane >> 4) * 2;
    const int row0 = tile * 16;

    const float* __restrict__ s1b  = s1 + (size_t)b * N_DIM + row0;
    const float* __restrict__ s2b  = s2 + (size_t)b * N_DIM;
    const int*   __restrict__ adjb = adj + ((size_t)b * N_DIM + row0) * N_DIM;
    const float* __restrict__ hb   = h + (size_t)b * N_DIM * F_OUTT;

    // hoist the 16 row scores into registers (uniform per block)
    float s1r[16];
    #pragma unroll
    for (int r = 0; r < 16; ++r) s1r[r] = s1b[r];

    float psum[16];
    #pragma unroll
    for (int r = 0; r < 16; ++r) psum[r] = 0.f;

    v8f acc = {};

    for (int j0 = 0; j0 < N_DIM; j0 += JB) {
        // stage h[j0:j0+JB][0:64] -> LDS (2048 float4, 16 per thread, coalesced)
        const float4* hsrc = (const float4*)(hb + (size_t)j0 * F_OUTT);
        float4* hdst = (float4*)h_s;
        #pragma unroll
        for (int i = 0; i < (JB * F_OUTT / 4) / 128; ++i)
            hdst[t + i * 128] = hsrc[t + i * 128];

        // attention weights: thread t handles column j0+t of each of 16 rows.
        // adj is read exactly once -> non-temporal; prefetch next chunk.
        const float s2v = s2b[j0 + t];
        const bool  pf  = (j0 + JB) < N_DIM;
        #pragma unroll
        for (int r = 0; r < 16; ++r) {
            const int aj = __builtin_nontemporal_load(
                &adjb[(size_t)r * N_DIM + j0 + t]);            // coalesced, TH=NT
            if (pf)
                __builtin_prefetch(&adjb[(size_t)r * N_DIM + j0 + JB + t], 0, 0);
            float e = s1r[r] + s2v;
            e = e > 0.f ? e : 0.f;                             // relu
            const float p = (aj > 0) ? __expf(e) : 0.f;        // masked exp
            att_s[r * ATT_LD + t] = p;
            psum[r] += p;
        }
        __syncthreads();

        // accumulate att_chunk(16xJB) @ h_chunk(JBx16w..) via WMMA f32 16x16x4
        #pragma unroll 8
        for (int kk = 0; kk < JB; kk += 4) {
            v2f a, bb;
            a.x  = att_s[m * ATT_LD + kk + kh];
            a.y  = att_s[m * ATT_LD + kk + kh + 1];
            bb.x = h_s[(kk + kh)     * F_OUTT + wave * 16 + m];
            bb.y = h_s[(kk + kh + 1) * F_OUTT + wave * 16 + m];
            acc = __builtin_amdgcn_wmma_f32_16x16x4_f32(false, a, false, bb,
                                                        (short)0, acc, false, false);
        }
        __syncthreads();
    }

    // reduce per-thread partial row sums (reuse att_s)
    #pragma unroll
    for (int r = 0; r < 16; ++r) att_s[r * ATT_LD + t] = psum[r];
    __syncthreads();
    if (t < 16) {
        float s = 0.f;
        for (int j = 0; j < JB; ++j) s += att_s[t * ATT_LD + j];
        rowsum_s[t] = s;
    }
    __syncthreads();

    // normalize, ELU, store (output written once -> non-temporal)
    float* outp = out + ((size_t)b * N_DIM + row0) * F_OUTT + wave * 16 + m;
    #pragma unroll
    for (int r = 0; r < 8; ++r) {
        const int rr = r + 8 * (lane >> 4);
        const float s = rowsum_s[rr];
        float v = (s > 0.f) ? acc[r] / s : 0.f;   // all-masked row -> 0 (matches ref)
        v = (v > 0.f) ? v : (__expf(v) - 1.f);    // elu, alpha=1
        __builtin_nontemporal_store(v, &outp[(size_t)rr * F_OUTT]);
    }
}

// ---------------------------------------------------------------------------
extern "C" void kernel_launch(void* const* d_in, const int* in_sizes, int n_in,
                              void* d_out, int out_size, void* d_ws, size_t ws_size,
                              hipStream_t stream) {
    const float* X   = (const float*)d_in[0];   // (B, N, 128)
    const float* W   = (const float*)d_in[1];   // (128, 64)
    const float* a   = (const float*)d_in[2];   // (128, 1)
    const int*   adj = (const int*)d_in[3];     // (B, N, N)
    // d_in[4] = N (scalar), compile-time constant here

    float* h  = (float*)d_ws;                              // B*N*64 floats (8 MB)
    float* s1 = h  + (size_t)B_DIM * N_DIM * F_OUTT;       // B*N floats
    float* s2 = s1 + (size_t)B_DIM * N_DIM;                // B*N floats

    gat_h_kernel<<<dim3(B_DIM * N_DIM / 16), dim3(128), 0, stream>>>(X, W, h);
    gat_scores_kernel<<<dim3((B_DIM * N_DIM + 255) / 256), dim3(256), 0, stream>>>(h, a, s1, s2);
    gat_attn_kernel<<<dim3(N_DIM / 16, B_DIM), dim3(128), 0, stream>>>(h, s1, s2, adj,
                                                                       (float*)d_out);
}